// ProjectedGraphEncoder_46170898432648
// MI455X (gfx1250) — compile-verified
//
#include <hip/hip_runtime.h>
#include <hip/hip_bf16.h>

typedef __attribute__((ext_vector_type(2))) float v2f;
typedef __attribute__((ext_vector_type(4))) float v4f;
typedef __attribute__((ext_vector_type(8))) float v8f;

#define CAND 64
#define DIM 128
#define WAVES_PER_WG 4
#define KV_ROW_DW 132                     // 128 dwords + 4 pad -> 528B row stride (16B aligned)
#define KV_DW (CAND * KV_ROW_DW)          // 8448 dwords
// per-wave LDS slice layout (dwords): [kv 8448][q 128][scores 64][attn 64][cand 64]
#define QOFF   KV_DW
#define SOFF   (QOFF + DIM)
#define AOFF   (SOFF + CAND)
#define COFF   (AOFF + CAND)
#define SLICE_DW (COFF + CAND)            // 8768 dwords = 35072 bytes per wave

__global__ __launch_bounds__(WAVES_PER_WG * 32)
void ragged_attn_kernel(const float* __restrict__ Xq,
                        const float* __restrict__ Xt,
                        const int* __restrict__ u_idx,
                        const int* __restrict__ cand_idx,
                        const int* __restrict__ cand_len,
                        float* __restrict__ out,
                        int n_nodes)
{
    extern __shared__ __align__(16) float smem[];
    const int lane = threadIdx.x & 31;
    const int wave = threadIdx.x >> 5;
    const int node = blockIdx.x * WAVES_PER_WG + wave;
    if (node >= n_nodes) return;

    float* sl  = smem + wave * SLICE_DW;
    float* kvs = sl;              // 64 x 132 dwords, padded KV tile
    float* qs  = sl + QOFF;       // 128 floats, query row
    float* ss  = sl + SOFF;       // 64 scores
    float* as  = sl + AOFF;       // 64 attention weights
    int*   cs  = (int*)(sl + COFF);

    // ---- stage candidate indices into LDS (coalesced) ----
    const size_t cbase = (size_t)node * CAND;
    cs[2 * lane]     = cand_idx[cbase + 2 * lane];
    cs[2 * lane + 1] = cand_idx[cbase + 2 * lane + 1];

    // ---- issue 64 async gathers: one KV row (512B) per instruction, 16B/lane ----
    const unsigned kv_lds_base = (unsigned)(size_t)kvs;   // flat LDS addr: low 32 bits = LDS offset
    const unsigned long long xt_base = (unsigned long long)(const void*)Xt;
    #pragma unroll 4
    for (int c = 0; c < CAND; ++c) {
        const int cand = cs[c];                            // uniform per wave
        const unsigned voff  = (unsigned)cand * (DIM * 4u) + (unsigned)lane * 16u;
        const unsigned laddr = kv_lds_base + (unsigned)c * (KV_ROW_DW * 4u) + (unsigned)lane * 16u;
        asm volatile("global_load_async_to_lds_b128 %0, %1, %2"
                     :: "v"(laddr), "v"(voff), "s"(xt_base) : "memory");
    }

    // ---- overlap: fetch query row and length while the gather is in flight ----
    const int u = u_idx[node];
    v4f qv = *(const v4f*)(Xq + (size_t)u * DIM + 4 * lane);
    *(v4f*)(qs + 4 * lane) = qv;
    const int len = cand_len[node];

    asm volatile("s_wait_asynccnt 0x0" ::: "memory");

    // ---- phase 1: scores = KV (64x128) . q  via V_WMMA_F32_16X16X4_F32 ----
    // A tile: 16 candidates x 4 k-dims from LDS.  B: q in column 0 only.
    // D column 0 (N=0) lives in lanes 0 (M=0..7) and 16 (M=8..15), VGPRs 0..7.
    v8f acc0 = {}, acc1 = {}, acc2 = {}, acc3 = {};
    const int halfsel = lane >> 4;     // 0: k+0/k+1, 1: k+2/k+3
    const int lrow    = lane & 15;
    #pragma unroll 8
    for (int k0 = 0; k0 < DIM; k0 += 4) {
        const int kk = k0 + halfsel * 2;
        float b0 = qs[kk];
        float b1 = qs[kk + 1];
        const bool sel = (lrow == 0);
        v2f bq;
        bq.x = sel ? b0 : 0.0f;
        bq.y = sel ? b1 : 0.0f;
        v2f a0 = *(v2f*)(kvs + ( 0 + lrow) * KV_ROW_DW + kk);
        v2f a1 = *(v2f*)(kvs + (16 + lrow) * KV_ROW_DW + kk);
        v2f a2 = *(v2f*)(kvs + (32 + lrow) * KV_ROW_DW + kk);
        v2f a3 = *(v2f*)(kvs + (48 + lrow) * KV_ROW_DW + kk);
        acc0 = __builtin_amdgcn_wmma_f32_16x16x4_f32(false, a0, false, bq, (short)0, acc0, false, false);
        acc1 = __builtin_amdgcn_wmma_f32_16x16x4_f32(false, a1, false, bq, (short)0, acc1, false, false);
        acc2 = __builtin_amdgcn_wmma_f32_16x16x4_f32(false, a2, false, bq, (short)0, acc2, false, false);
        acc3 = __builtin_amdgcn_wmma_f32_16x16x4_f32(false, a3, false, bq, (short)0, acc3, false, false);
    }

    // ---- spill column 0 of D to the score buffer (lanes 0 and 16 hold it) ----
    if (lrow == 0) {
        const int off = halfsel * 8;
        #pragma unroll
        for (int r = 0; r < 8; ++r) {
            ss[ 0 + off + r] = acc0[r];
            ss[16 + off + r] = acc1[r];
            ss[32 + off + r] = acc2[r];
            ss[48 + off + r] = acc3[r];
        }
    }

    // ---- softmax over 64 scores (2 per lane), wave32 reductions ----
    const float scale = 0.08838834764831845f;   // 1/sqrt(128)
    float s0 = ss[lane] * scale;
    float s1 = ss[lane + 32] * scale;
    if (lane      >= len) s0 = -1e30f;
    if (lane + 32 >= len) s1 = -1e30f;
    float m = fmaxf(s0, s1);
    #pragma unroll
    for (int off = 16; off > 0; off >>= 1) m = fmaxf(m, __shfl_xor(m, off, 32));
    float e0 = __expf(s0 - m);
    float e1 = __expf(s1 - m);
    float t = e0 + e1;
    #pragma unroll
    for (int off = 16; off > 0; off >>= 1) t += __shfl_xor(t, off, 32);
    const float inv = 1.0f / t;
    as[lane]      = e0 * inv;
    as[lane + 32] = e1 * inv;

    // ---- phase 2: out = attn^T . KV  (each lane owns 4 contiguous d-columns) ----
    v4f o = {};
    #pragma unroll 8
    for (int c = 0; c < CAND; ++c) {
        const float a = as[c];                               // LDS broadcast
        v4f kvv = *(v4f*)(kvs + c * KV_ROW_DW + 4 * lane);   // ds_load_b128
        o += a * kvv;
    }
    *(v4f*)(out + (size_t)node * DIM + 4 * lane) = o;        // global_store_b128, coalesced
}

extern "C" void kernel_launch(void* const* d_in, const int* in_sizes, int n_in,
                              void* d_out, int out_size, void* d_ws, size_t ws_size,
                              hipStream_t stream) {
    const float* Xq       = (const float*)d_in[0];
    const float* Xt       = (const float*)d_in[1];
    const int*   u_idx    = (const int*)d_in[2];
    const int*   cand_idx = (const int*)d_in[3];
    const int*   cand_len = (const int*)d_in[4];
    float*       out      = (float*)d_out;

    const int n_nodes = in_sizes[2];                 // N = 32768
    const int blocks  = (n_nodes + WAVES_PER_WG - 1) / WAVES_PER_WG;
    const size_t shmem = (size_t)WAVES_PER_WG * SLICE_DW * sizeof(float);  // ~137 KB of 320 KB WGP LDS

    hipLaunchKernelGGL(ragged_attn_kernel, dim3(blocks), dim3(WAVES_PER_WG * 32), shmem, stream,
                       Xq, Xt, u_idx, cand_idx, cand_len, out, n_nodes);
}